// GCN_71124658421835
// MI455X (gfx1250) — compile-verified
//
#include <hip/hip_runtime.h>

typedef float v2f __attribute__((ext_vector_type(2)));
typedef float v8f __attribute__((ext_vector_type(8)));

// Native fp32 global atomic add, pinned via CDNA5 inline asm (ISA §15.18.3):
//   global_atomic_add_f32 vaddr[2], vdata, off   (non-returning, TH=RT)
// The sums here are order-insensitive, so relaxed native atomics are exact.
__device__ __forceinline__ void atomAddF(float* p, float v) {
  asm volatile("global_atomic_add_f32 %0, %1, off" : : "v"(p), "v"(v) : "memory");
}

// ---------------- zero fill ----------------
__global__ void k_fill0(float* __restrict__ p, long long n) {
  long long i = (long long)blockIdx.x * blockDim.x + threadIdx.x;
  if (i < n) p[i] = 0.0f;
}

// ---------------- degree count (atomic f32 add of 1.0 per edge; counts < 2^24 are exact) ----
__global__ void k_deg(const int* __restrict__ dst, float* __restrict__ deg, int E) {
  int e = blockIdx.x * blockDim.x + threadIdx.x;
  if (e < E) atomAddF(deg + dst[e], 1.0f);
}

// ---------------- dinv = rsqrt(deg + 1) in place ----------------
__global__ void k_dinv(float* __restrict__ d, int N) {
  int i = blockIdx.x * blockDim.x + threadIdx.x;
  if (i < N) d[i] = rsqrtf(d[i] + 1.0f);
}

// ---------------- fp32 WMMA GEMM: C[M,N] = A[M,K] * B[K,N] ----------------
// One wave per 16x16 output tile, K-loop in steps of 4 using V_WMMA_F32_16X16X4_F32.
// K, N compile-time -> all load/store offsets fold into immediate offsets.
// M must be a multiple of 16 (50000 = 3125*16): no bounds guards needed.
// ISA layouts (CDNA5 ISA):
//   A 16x4 : lanes 0-15 -> M=lane, VGPR0=K0, VGPR1=K1 ; lanes 16-31 -> M=lane-16, K2/K3
//   B 4x16 : lanes 0-15 -> N=lane, VGPR0=K0 row, VGPR1=K1 ; lanes 16-31 -> K2/K3
//   D 16x16: VGPR r -> row (r + 8*half), col = lane&15
template <int K, int N>
__launch_bounds__(32)
__global__ void k_gemm_wmma(const float* __restrict__ A, const float* __restrict__ B,
                            float* __restrict__ C) {
  const int lane = threadIdx.x;      // 0..31 (wave32)
  const int half = lane >> 4;        // 0 or 1
  const int lm   = lane & 15;
  const int m0 = blockIdx.x * 16;
  const int n0 = blockIdx.y * 16;

  const float* __restrict__ arow = A + (long long)(m0 + lm) * K + 2 * half;  // 8B aligned
  const float* __restrict__ bp   = B + (long long)(2 * half) * N + n0 + lm;

  v8f acc = {};
#pragma unroll 8
  for (int k = 0; k < K; k += 4) {
    v2f a = *(const v2f*)(arow + k);       // A[row][k+2h], A[row][k+2h+1]
    v2f b;
    b.x = bp[k * N];                       // B[k+2h  ][n0+lm]  (const offsets)
    b.y = bp[k * N + N];                   // B[k+2h+1][n0+lm]
    acc = __builtin_amdgcn_wmma_f32_16x16x4_f32(
        /*neg_a=*/false, a, /*neg_b=*/false, b,
        /*c_mod=*/(short)0, acc, /*reuse_a=*/false, /*reuse_b=*/false);
  }

  float* __restrict__ cp = C + (long long)(m0 + 8 * half) * N + n0 + lm;
#pragma unroll
  for (int r = 0; r < 8; ++r)
    cp[r * N] = acc[r];                    // constant row offsets, no guards
}

// ---------------- edge scatter, F=64: one wave per edge, float2 per lane ----------------
// wid forced wave-uniform via readfirstlane -> src/dst/dinv become scalar broadcast loads.
__launch_bounds__(256)
__global__ void k_scatter64(const float* __restrict__ h, const float* __restrict__ dinv,
                            const int* __restrict__ src, const int* __restrict__ dst,
                            float* __restrict__ agg, int E) {
  int wid  = blockIdx.x * (blockDim.x >> 5) + (threadIdx.x >> 5);
  wid = __builtin_amdgcn_readfirstlane(wid);
  int lane = threadIdx.x & 31;
  if (wid >= E) return;
  int s = src[wid];
  int d = dst[wid];
  float w = dinv[s];
  float2 v = *((const float2*)(h + (long long)s * 64) + lane);
  float* ap = agg + (long long)d * 64 + lane * 2;
  atomAddF(ap,     v.x * w);
  atomAddF(ap + 1, v.y * w);
}

// ---------------- edge scatter, F=128: one wave per edge, float4 per lane ----------------
__launch_bounds__(256)
__global__ void k_scatter128(const float* __restrict__ h, const float* __restrict__ dinv,
                             const int* __restrict__ src, const int* __restrict__ dst,
                             float* __restrict__ agg, int E) {
  int wid  = blockIdx.x * (blockDim.x >> 5) + (threadIdx.x >> 5);
  wid = __builtin_amdgcn_readfirstlane(wid);
  int lane = threadIdx.x & 31;
  if (wid >= E) return;
  int s = src[wid];
  int d = dst[wid];
  float w = dinv[s];
  float4 v = *((const float4*)(h + (long long)s * 128) + lane);
  float* ap = agg + (long long)d * 128 + lane * 4;
  atomAddF(ap,     v.x * w);
  atomAddF(ap + 1, v.y * w);
  atomAddF(ap + 2, v.z * w);
  atomAddF(ap + 3, v.w * w);
}

// ---------------- finish: out = [relu](dinv*agg + dinv^2*h + b), float4 granularity ----------------
template <int F, bool RELU>
__launch_bounds__(256)
__global__ void k_finish(const float* __restrict__ h, const float* __restrict__ agg,
                         const float* __restrict__ dinv, const float* __restrict__ bias,
                         float* __restrict__ out, int N) {
  int idx = blockIdx.x * blockDim.x + threadIdx.x;     // one float4 per thread
  const int F4 = F / 4;
  long long total = (long long)N * F4;
  if (idx >= total) return;
  int i  = idx / F4;                                    // power of two -> shifts
  int f4 = idx - i * F4;
  float di = dinv[i];
  float s  = di * di;
  float4 a  = ((const float4*)agg)[idx];
  float4 hv = ((const float4*)h)[idx];
  float4 bv = ((const float4*)bias)[f4];
  float4 r;
  r.x = di * a.x + s * hv.x + bv.x;
  r.y = di * a.y + s * hv.y + bv.y;
  r.z = di * a.z + s * hv.z + bv.z;
  r.w = di * a.w + s * hv.w + bv.w;
  if (RELU) {
    r.x = fmaxf(r.x, 0.0f); r.y = fmaxf(r.y, 0.0f);
    r.z = fmaxf(r.z, 0.0f); r.w = fmaxf(r.w, 0.0f);
  }
  ((float4*)out)[idx] = r;
}

// ---------------- host orchestration ----------------
extern "C" void kernel_launch(void* const* d_in, const int* in_sizes, int n_in,
                              void* d_out, int out_size, void* d_ws, size_t ws_size,
                              hipStream_t stream) {
  const float* x  = (const float*)d_in[0];   // [N, 256]
  const int*   ei = (const int*)d_in[1];     // [2, E]
  const float* W1 = (const float*)d_in[2];   // [256, 64]
  const float* b1 = (const float*)d_in[3];   // [64]
  const float* W2 = (const float*)d_in[4];   // [64, 128]
  const float* b2 = (const float*)d_in[5];   // [128]

  const int FIN = 256, H = 64, FOUT = 128;
  const int N = in_sizes[0] / FIN;
  const int E = in_sizes[1] / 2;
  const int* srcI = ei;
  const int* dstI = ei + E;

  // Workspace layout (floats): [dinv N][h1 64N][agg1 64N][y1 64N][agg2 128N]
  // h2 (128N) aliases [h1|agg1] which are dead after finish1.
  float* ws   = (float*)d_ws;
  float* dinv = ws;
  float* h1   = dinv + (long long)N;
  float* agg1 = h1   + (long long)N * H;
  float* y1   = agg1 + (long long)N * H;
  float* agg2 = y1   + (long long)N * H;
  float* h2   = h1;   // alias

  const int TB = 256;
  auto blocks = [](long long n, int tb) { return (unsigned)((n + tb - 1) / tb); };

  // 1) zero accumulators
  k_fill0<<<blocks((long long)N, TB), TB, 0, stream>>>(dinv, N);
  k_fill0<<<blocks((long long)N * H, TB), TB, 0, stream>>>(agg1, (long long)N * H);
  k_fill0<<<blocks((long long)N * FOUT, TB), TB, 0, stream>>>(agg2, (long long)N * FOUT);

  // 2) degree + dinv
  k_deg<<<blocks(E, TB), TB, 0, stream>>>(dstI, dinv, E);
  k_dinv<<<blocks(N, TB), TB, 0, stream>>>(dinv, N);

  // 3) layer 1: GEMM (fp32 WMMA), scatter, finish (+ReLU)
  {
    dim3 g((unsigned)(N / 16), H / 16);
    k_gemm_wmma<256, 64><<<g, 32, 0, stream>>>(x, W1, h1);
  }
  k_scatter64<<<blocks((long long)E * 32, TB), TB, 0, stream>>>(h1, dinv, srcI, dstI, agg1, E);
  k_finish<64, true><<<blocks((long long)N * H / 4, TB), TB, 0, stream>>>(h1, agg1, dinv, b1, y1, N);

  // 4) layer 2: GEMM, scatter, finish (no ReLU) -> d_out
  {
    dim3 g((unsigned)(N / 16), FOUT / 16);
    k_gemm_wmma<64, 128><<<g, 32, 0, stream>>>(y1, W2, h2);
  }
  k_scatter128<<<blocks((long long)E * 32, TB), TB, 0, stream>>>(h2, dinv, srcI, dstI, agg2, E);
  k_finish<128, false><<<blocks((long long)N * FOUT / 4, TB), TB, 0, stream>>>(
      h2, agg2, dinv, b2, (float*)d_out, N);
}